// DiffDTW_12146167513178
// MI455X (gfx1250) — compile-verified
//
#include <hip/hip_runtime.h>
#include <cstddef>

typedef float v2f __attribute__((ext_vector_type(2)));
typedef float v8f __attribute__((ext_vector_type(8)));

constexpr int Bn = 128;   // batch
constexpr int Nn = 384;   // rows of A / DP rows
constexpr int Mn = 384;   // rows of B / DP cols
constexpr int Dd = 128;   // feature dim
constexpr float INFV = 1000000.0f;  // reference pseudo-infinity

// ---------------------------------------------------------------------------
// Kernel 0: squared norms of every row of seq_a and seq_b.
// One wave32 per row: lane loads float4 (coalesced 512B/wave), shuffle-reduce.
// ---------------------------------------------------------------------------
__global__ __launch_bounds__(256) void sqnorm_kernel(
    const float* __restrict__ A, const float* __restrict__ Bm,
    float* __restrict__ a2, float* __restrict__ b2) {
  const int wave = threadIdx.x >> 5;
  const int lane = threadIdx.x & 31;
  const int row  = blockIdx.x * 8 + wave;          // 0 .. 2*Bn*Nn-1
  const int nA = Bn * Nn;
  const float* src = (row < nA) ? (A + (size_t)row * Dd)
                                : (Bm + (size_t)(row - nA) * Dd);
  const float4* s4 = (const float4*)src;
  float4 v = s4[lane];                              // 32 lanes * 4 = 128
  float ssum = v.x * v.x + v.y * v.y + v.z * v.z + v.w * v.w;
#pragma unroll
  for (int m = 16; m >= 1; m >>= 1) ssum += __shfl_xor(ssum, m, 32);
  if (lane == 0) {
    if (row < nA) a2[row] = ssum;
    else          b2[row - nA] = ssum;
  }
}

// ---------------------------------------------------------------------------
// Kernel 1: D[b,i,j] = a2[b,i] + b2[b,j] - 2 * dot(a_i, b_j)
// Grid: (Nn/16, Bn).  Block: 256 threads = 8 waves.
// Each wave owns column tiles {w, w+8, w+16}; A fragment reused across tiles.
// V_WMMA_F32_16X16X4_F32: A is 16x4 f32 (v2f/lane), B is 4x16 (v2f/lane),
// C/D is 16x16 f32 (v8f/lane).
//   A frag: lane L (L<16): row i0+L, K = kk+{0,1}; L>=16: row i0+L-16, K=kk+{2,3}
//   B frag (Bt[k][n] = seq_b[j0+n][k]): identical per-lane addressing on seq_b
//   C/D:    element v of lane L -> row i0 + v + 8*(L/16), col j0 + (L%16)
// ---------------------------------------------------------------------------
__global__ __launch_bounds__(256) void pairdist_wmma_kernel(
    const float* __restrict__ A, const float* __restrict__ Bm,
    const float* __restrict__ a2, const float* __restrict__ b2,
    float* __restrict__ Dout) {
  const int b    = blockIdx.y;
  const int i0   = blockIdx.x * 16;
  const int wave = threadIdx.x >> 5;
  const int lane = threadIdx.x & 31;
  const int mrow = lane & 15;
  const int koff = (lane >> 4) << 1;                // 0 or 2

  const int j0_0 = (wave + 0)  * 16;
  const int j0_1 = (wave + 8)  * 16;
  const int j0_2 = (wave + 16) * 16;

  const float* arow  = A  + ((size_t)b * Nn + i0   + mrow) * Dd + koff;
  const float* brow0 = Bm + ((size_t)b * Mn + j0_0 + mrow) * Dd + koff;
  const float* brow1 = Bm + ((size_t)b * Mn + j0_1 + mrow) * Dd + koff;
  const float* brow2 = Bm + ((size_t)b * Mn + j0_2 + mrow) * Dd + koff;

  v8f acc0 = {}; v8f acc1 = {}; v8f acc2 = {};
#pragma unroll 4
  for (int kk = 0; kk < Dd; kk += 4) {
    v2f af  = *(const v2f*)(arow  + kk);
    v2f bf0 = *(const v2f*)(brow0 + kk);
    v2f bf1 = *(const v2f*)(brow1 + kk);
    v2f bf2 = *(const v2f*)(brow2 + kk);
    acc0 = __builtin_amdgcn_wmma_f32_16x16x4_f32(false, af, false, bf0,
                                                 (short)0, acc0, false, false);
    acc1 = __builtin_amdgcn_wmma_f32_16x16x4_f32(false, af, false, bf1,
                                                 (short)0, acc1, false, false);
    acc2 = __builtin_amdgcn_wmma_f32_16x16x4_f32(false, af, false, bf2,
                                                 (short)0, acc2, false, false);
  }

  // Epilogue: a2 + b2 - 2*dot, scatter per C/D layout.
  const int roff = (lane >> 4) << 3;                // 0 or 8
  const float* a2p = a2 + (size_t)b * Nn + i0 + roff;
  float a2v[8];
#pragma unroll
  for (int v = 0; v < 8; ++v) a2v[v] = a2p[v];
  const float b2v0 = b2[(size_t)b * Mn + j0_0 + mrow];
  const float b2v1 = b2[(size_t)b * Mn + j0_1 + mrow];
  const float b2v2 = b2[(size_t)b * Mn + j0_2 + mrow];

  float* dbase = Dout + ((size_t)b * Nn + i0 + roff) * Mn;
#pragma unroll
  for (int v = 0; v < 8; ++v) {
    float* drow = dbase + (size_t)v * Mn;
    drow[j0_0 + mrow] = a2v[v] + b2v0 - 2.0f * acc0[v];
    drow[j0_1 + mrow] = a2v[v] + b2v1 - 2.0f * acc1[v];
    drow[j0_2 + mrow] = a2v[v] + b2v2 - 2.0f * acc2[v];
  }
}

// ---------------------------------------------------------------------------
// Kernel 2: soft-DTW wavefront DP, one workgroup per batch, 384 threads.
// Three rotating anti-diagonal buffers in LDS, indexed by row i:
//   buf[s%3][i] = R[i][s-i].  Cell (i,j), s=i+j needs:
//   diag = buf[(s-2)%3][i-1], up = buf[(s-1)%3][i-1], left = buf[(s-1)%3][i].
// Boundary entries (j==0 and i==0) are kept at INFV; R[0][0]=0 seeds s=0.
// gamma = 1 folded into the stable softmin.
// ---------------------------------------------------------------------------
__global__ __launch_bounds__(Nn) void softdtw_dp_kernel(
    const float* __restrict__ Dmat, float* __restrict__ out) {
  const int b = blockIdx.x;
  const int t = threadIdx.x;       // 0..383
  const int i = t + 1;             // DP row 1..384

  __shared__ float buf[3][Nn + 4];
  float* flat = &buf[0][0];
  for (int idx = t; idx < 3 * (Nn + 4); idx += blockDim.x) flat[idx] = INFV;
  __syncthreads();
  if (t == 0) buf[0][0] = 0.0f;    // R[0][0]; buf slot 0 == diag s=0
  __syncthreads();

  const float* drow = Dmat + ((size_t)b * Nn + (i - 1)) * Mn;

  // rotating slot ids: c0 = s%3, c1 = (s-1)%3, c2 = (s-2)%3 ; start at s=2
  int c0 = 2, c1 = 1, c2 = 0;

  // prefetch D for s=2 (clamped; only i=1 is active there)
  int pidx = 2 - i - 1; pidx = pidx < 0 ? 0 : (pidx > Mn - 1 ? Mn - 1 : pidx);
  float dcur = drow[pidx];

  for (int s = 2; s <= Nn + Mn; ++s) {
    const int j = s - i;

    // prefetch D for next step before this step's barrier
    int nidx = s - i; nidx = nidx < 0 ? 0 : (nidx > Mn - 1 ? Mn - 1 : nidx);
    const float dnext = drow[nidx];

    float val = INFV;
    if (j >= 1 && j <= Mn) {
      const float diag = buf[c2][i - 1];
      const float up   = buf[c1][i - 1];
      const float left = buf[c1][i];
      const float mn   = fminf(diag, fminf(up, left));
      const float ssum = __expf(mn - diag) + __expf(mn - up) + __expf(mn - left);
      val = dcur + mn - __logf(ssum);      // gamma = 1
    }
    if (j >= 0 && j <= Mn) buf[c0][i] = (j == 0) ? INFV : val;  // j==0 -> R[i][0]=INF
    if (t == 0) buf[c0][0] = INFV;                              // R[0][s]=INF (s>0)
    __syncthreads();

    dcur = dnext;
    const int tmp = c2; c2 = c1; c1 = c0; c0 = tmp;             // advance s
  }

  // final diag s = 768 lives in slot (768 % 3) == 0 at index i = 384
  if (t == 0) out[b] = buf[0][Mn];
}

// ---------------------------------------------------------------------------
extern "C" void kernel_launch(void* const* d_in, const int* in_sizes, int n_in,
                              void* d_out, int out_size, void* d_ws, size_t ws_size,
                              hipStream_t stream) {
  (void)in_sizes; (void)n_in; (void)out_size; (void)ws_size;
  const float* seq_a = (const float*)d_in[0];
  const float* seq_b = (const float*)d_in[1];
  float* out = (float*)d_out;

  float* ws = (float*)d_ws;
  float* a2 = ws;                               // Bn*Nn
  float* b2 = ws + (size_t)Bn * Nn;             // Bn*Mn
  float* D  = ws + (size_t)Bn * Nn + (size_t)Bn * Mn;  // Bn*Nn*Mn

  // Kernel 0: row norms (2*128*384 rows, 8 waves/block)
  sqnorm_kernel<<<(2 * Bn * Nn) / 8, 256, 0, stream>>>(seq_a, seq_b, a2, b2);

  // Kernel 1: WMMA pairwise distances
  dim3 g1(Nn / 16, Bn);
  pairdist_wmma_kernel<<<g1, 256, 0, stream>>>(seq_a, seq_b, a2, b2, D);

  // Kernel 2: wavefront soft-DTW, one workgroup per batch
  softdtw_dp_kernel<<<Bn, Nn, 0, stream>>>(D, out);
}